// Chamfer_Density_Loss_88278757802043
// MI455X (gfx1250) — compile-verified
//
#include <hip/hip_runtime.h>
#include <hip/hip_bf16.h>
#include <math.h>

typedef __attribute__((ext_vector_type(2))) float v2f;
typedef __attribute__((ext_vector_type(8))) float v8f;

constexpr int Bb = 8;      // batches
constexpr int Nn = 4096;   // gts points
constexpr int Mm = 4096;   // pred points
constexpr int Kk = 16;     // density_k
constexpr int STR = 136;   // padded LDS stride (128 cols + 8 pad) to avoid bank conflicts
constexpr int NBLK = Bb * (Nn / 16);  // 2048 row-tile workgroups

// Monotone float -> uint key (valid for negatives too: slight negative dists
// can appear from rx+ry-2xy rounding).
__device__ __forceinline__ unsigned int fkey(float f) {
  unsigned int u = __float_as_uint(f);
  return (u & 0x80000000u) ? ~u : (u ^ 0x80000000u);
}
__device__ __forceinline__ float funkey(unsigned int k) {
  unsigned int u = (k & 0x80000000u) ? (k ^ 0x80000000u) : ~k;
  return __uint_as_float(u);
}

// Merge two sorted 16-lists, keep smallest 16 (two-pointer).
__device__ __forceinline__ void mergeStep(const float* S, float* D, int st, int row, int sub) {
  if ((sub & (2 * st - 1)) == 0) {
    const float* L1 = &S[(row * 16 + sub) * 16];
    const float* L2 = &S[(row * 16 + sub + st) * 16];
    float* O = &D[(row * 16 + sub) * 16];
    int i = 0, j = 0;
    for (int k = 0; k < 16; ++k) {
      float av = L1[i], bv = L2[j];
      if (av <= bv) { O[k] = av; ++i; } else { O[k] = bv; ++j; }
    }
  }
}

template <bool PASS2>
__global__ __launch_bounds__(256) void cd_pass(
    const float* __restrict__ gts, const float* __restrict__ other,
    unsigned int* __restrict__ colmin, float* __restrict__ val2ws,
    float* __restrict__ partials) {
  __shared__ float gcoord[16][3];
  __shared__ float rxl[16];
  __shared__ float dist[16 * STR];      // raw (-2 x.y + ry) chunk: 16 rows x 128 cols
  __shared__ float MA[16 * 16 * 16];    // merge ping
  __shared__ float MB[16 * 16 * 16];    // merge pong
  __shared__ float red[256];
  __shared__ float red2[256];

  const int tid = threadIdx.x;
  const int b = blockIdx.x / (Nn / 16);
  const int n0 = (blockIdx.x % (Nn / 16)) * 16;

  if (tid < 16) {
    const float* p = gts + ((size_t)b * Nn + n0 + tid) * 3;
    float x = p[0], y = p[1], z = p[2];
    gcoord[tid][0] = x; gcoord[tid][1] = y; gcoord[tid][2] = z;
    rxl[tid] = x * x + y * y + z * z;
  }
  __syncthreads();

  const int lane = tid & 31;
  const int wave = tid >> 5;
  const int L = lane & 15;
  const int hi = lane >> 4;

  // A operand (16x4 f32): row L = (x0,x1,x2,1); lo lanes hold K=0,1; hi lanes K=2,3
  v2f a;
  a.x = hi ? gcoord[L][2] : gcoord[L][0];
  a.y = hi ? 1.0f : gcoord[L][1];

  // selection mapping: 16 threads per row
  const int row = tid >> 4;
  const int sub = tid & 15;

  float topv[16];
#pragma unroll
  for (int i = 0; i < 16; ++i) topv[i] = INFINITY;

  for (int chunk = 0; chunk < Mm / 128; ++chunk) {
    const int ct = chunk * 8 + wave;
    const int col = ct * 16 + L;
    const float* q = other + ((size_t)b * Mm + col) * 3;
    float q0 = q[0], q1 = q[1], q2 = q[2];
    float ry = q0 * q0 + q1 * q1 + q2 * q2;
    // B operand (4x16 f32): col = (-2y0,-2y1,-2y2,ry); lo lanes K=0,1; hi lanes K=2,3
    v2f bb;
    bb.x = hi ? (-2.0f * q2) : (-2.0f * q0);
    bb.y = hi ? ry : (-2.0f * q1);
    v8f c = {0.f, 0.f, 0.f, 0.f, 0.f, 0.f, 0.f, 0.f};
    v8f d = __builtin_amdgcn_wmma_f32_16x16x4_f32(false, a, false, bb, (short)0, c,
                                                  false, false);
    if (PASS2) {
      // column min over the 16 rows of this tile (rx added per row)
      float m = INFINITY;
#pragma unroll
      for (int v = 0; v < 8; ++v) m = fminf(m, d[v] + rxl[v + hi * 8]);
      m = fminf(m, __shfl_xor(m, 16, 32));
      if (!hi) atomicMin(&colmin[(size_t)b * Mm + col], fkey(m));
    }
    // stash raw values (row ordering unaffected by +rx)
#pragma unroll
    for (int v = 0; v < 8; ++v) dist[(v + hi * 8) * STR + wave * 16 + L] = d[v];
    __syncthreads();

    // each thread scans 8 of the 128 columns for its row
#pragma unroll
    for (int j = 0; j < 8; ++j) {
      float cval = dist[row * STR + sub + 16 * j];
      if (cval < topv[15]) {
        float x = cval;
#pragma unroll
        for (int i = 0; i < 16; ++i) {
          float lo = fminf(topv[i], x);
          x = fmaxf(topv[i], x);
          topv[i] = lo;
        }
      }
    }
    __syncthreads();
  }

  // merge 16 sorted lists per row -> global ascending top-16
#pragma unroll
  for (int i = 0; i < 16; ++i) MA[(row * 16 + sub) * 16 + i] = topv[i];
  __syncthreads();
  mergeStep(MA, MB, 1, row, sub); __syncthreads();
  mergeStep(MB, MA, 2, row, sub); __syncthreads();
  mergeStep(MA, MB, 4, row, sub); __syncthreads();
  mergeStep(MB, MA, 8, row, sub); __syncthreads();

  // final list for row r: MA[(r*16+0)*16 + k]; add back rx
  float myval = MA[row * 256 + sub] + rxl[row];
  const size_t vidx = ((size_t)b * Nn + n0 + row) * Kk + sub;

  if (!PASS2) {
    val2ws[vidx] = myval;  // sorted k-NN dists of gts vs gts
  } else {
    float v2v = val2ws[vidx];
    float df = myval - v2v;
    red[tid] = df * df;                         // density partial
    red2[tid] = (sub == 0) ? myval : 0.0f;      // row min = smallest element
    __syncthreads();
    for (int s = 128; s > 0; s >>= 1) {
      if (tid < s) { red[tid] += red[tid + s]; red2[tid] += red2[tid + s]; }
      __syncthreads();
    }
    if (tid == 0) {
      partials[blockIdx.x] = red2[0];           // rowmin sums
      partials[NBLK + blockIdx.x] = red[0];     // density sums
    }
  }
}

__global__ void cd_init(unsigned int* colmin, int n) {
  int i = blockIdx.x * blockDim.x + threadIdx.x;
  if (i < n) colmin[i] = 0xFFFFFFFFu;
}

__global__ __launch_bounds__(256) void cd_final(const unsigned int* __restrict__ colmin,
                                                const float* __restrict__ partials,
                                                float* __restrict__ out) {
  __shared__ float r0[256], r1[256], r2[256];
  int tid = threadIdx.x;
  float s0 = 0.f, s1 = 0.f, s2 = 0.f;
  for (int i = tid; i < Bb * Mm; i += 256) s0 += funkey(colmin[i]);
  for (int i = tid; i < NBLK; i += 256) { s1 += partials[i]; s2 += partials[NBLK + i]; }
  r0[tid] = s0; r1[tid] = s1; r2[tid] = s2;
  __syncthreads();
  for (int s = 128; s > 0; s >>= 1) {
    if (tid < s) { r0[tid] += r0[tid + s]; r1[tid] += r1[tid + s]; r2[tid] += r2[tid + s]; }
    __syncthreads();
  }
  if (tid == 0) {
    float loss1 = r0[0] / (float)(Bb * Mm);   // mean of col mins
    float loss2 = r1[0] / (float)(Bb * Nn);   // mean of row mins
    out[0] = loss1 + loss2;                   // chamfer
    out[1] = r2[0] / (float)(Bb * Nn * Kk);   // density
  }
}

extern "C" void kernel_launch(void* const* d_in, const int* in_sizes, int n_in,
                              void* d_out, int out_size, void* d_ws, size_t ws_size,
                              hipStream_t stream) {
  (void)in_sizes; (void)n_in; (void)out_size; (void)ws_size;
  const float* gts = (const float*)d_in[0];
  const float* preds = (const float*)d_in[1];
  float* out = (float*)d_out;

  char* ws = (char*)d_ws;
  unsigned int* colmin = (unsigned int*)ws;                       // B*M uints
  float* val2 = (float*)(ws + (size_t)Bb * Mm * sizeof(unsigned int));
  float* partials = (float*)(ws + (size_t)Bb * Mm * sizeof(unsigned int)
                                + (size_t)Bb * Nn * Kk * sizeof(float));  // 2*NBLK

  cd_init<<<(Bb * Mm + 255) / 256, 256, 0, stream>>>(colmin, Bb * Mm);
  dim3 grid(NBLK);
  cd_pass<false><<<grid, 256, 0, stream>>>(gts, gts, colmin, val2, partials);
  cd_pass<true><<<grid, 256, 0, stream>>>(gts, preds, colmin, val2, partials);
  cd_final<<<1, 256, 0, stream>>>(colmin, partials, out);
}